// GlobalFusion_18107582120748
// MI455X (gfx1250) — compile-verified
//
#include <hip/hip_runtime.h>
#include <stdint.h>

// ---------------------------------------------------------------------------
// GlobalFusion: voxel-hash gather + add.
//   out[i,:] = local[i,:] + (valid ? global[map[(lc[i]+lb)>>3 + gb]] : 0)
// Memory-bound (AI ~0.12 flop/byte; roofline ~18us @ 23.3 TB/s). Strategy:
//   - 32^3 int map lives in d_ws (128 KB)
//   - main kernel: 256 thr/block (8 wave32), 4 rows/block, float4 per thread
//   - streaming local_features staged via async global->LDS
//     (GLOBAL_LOAD_ASYNC_TO_LDS_B128, ASYNCcnt) with TH=NT so the 205 MB
//     stream does not thrash L2
//   - out written with nontemporal (TH=NT) b128 stores
//   - global_features gathered with default RT policy -> its 20 MB working
//     set stays resident in the 192 MB L2, so gathers never go to HBM
// ---------------------------------------------------------------------------

#define GRID        32
#define GRID3       (GRID * GRID * GRID)
#define SCALE_SHIFT 3          // SCALE_RATIO = 8
#define C           256
#define C4          64         // float4 lanes per row
#define ROWS_PB     4
#define THREADS     (ROWS_PB * C4)   // 256

#if __has_builtin(__builtin_amdgcn_global_load_async_to_lds_b128)
#define HAVE_ASYNC 1
#else
#define HAVE_ASYNC 0
#endif

typedef int   v4i __attribute__((ext_vector_type(4)));
typedef float v4f __attribute__((ext_vector_type(4)));

#if HAVE_ASYNC
typedef __attribute__((address_space(1))) v4i as1_v4i;   // global
typedef __attribute__((address_space(3))) v4i as3_v4i;   // LDS
#endif

__device__ __forceinline__ void stage16_nt(const void* gsrc, void* ldst) {
#if HAVE_ASYNC
    // async 16B global -> LDS, tracked by ASYNCcnt, no VGPR data path.
    // cpol=1: TH=NT (non-temporal) -- streaming read, keep it out of L2.
    __builtin_amdgcn_global_load_async_to_lds_b128(
        (as1_v4i*)(uintptr_t)gsrc, (as3_v4i*)ldst, /*offset=*/0, /*cpol=*/1);
#else
    *(v4f*)ldst = __builtin_nontemporal_load((const v4f*)gsrc);
#endif
}

__device__ __forceinline__ void wait_async() {
#if HAVE_ASYNC
#if __has_builtin(__builtin_amdgcn_s_wait_asynccnt)
    __builtin_amdgcn_s_wait_asynccnt(0);
#else
    asm volatile("s_wait_asynccnt 0" ::: "memory");
#endif
#endif
}

// ---- kernel 1: map = -1 ----------------------------------------------------
__global__ void gf_init_map(int* __restrict__ map) {
    int i = blockIdx.x * blockDim.x + threadIdx.x;
    if (i < GRID3) map[i] = -1;
}

// ---- kernel 2: scatter arange(Ng) by global_coords -------------------------
__global__ void gf_scatter_map(const int* __restrict__ gcoord,
                               int* __restrict__ map, int Ng) {
    int i = blockIdx.x * blockDim.x + threadIdx.x;
    if (i >= Ng) return;
    int x = gcoord[i * 3 + 0];
    int y = gcoord[i * 3 + 1];
    int z = gcoord[i * 3 + 2];
    map[(x * GRID + y) * GRID + z] = i;   // duplicate coords: last writer wins
}

// ---- kernel 3: fused gather+add -------------------------------------------
__global__ __launch_bounds__(THREADS) void gf_fuse(
    const float* __restrict__ lf,   // [Nl, 256]
    const int*   __restrict__ lc,   // [Nl, 3]
    const int*   __restrict__ lb,   // [3]
    const float* __restrict__ gfeat,// [Ng, 256]
    const int*   __restrict__ gb,   // [3]
    const int*   __restrict__ map,  // [32^3]
    float*       __restrict__ out,  // [Nl, 256]
    int Nl) {
    __shared__ float tile[THREADS * 4];          // 4 KB: 4 rows x 256 floats

    const int tid = threadIdx.x;
    const int r   = tid >> 6;                    // row within block
    const int c   = tid & (C4 - 1);              // float4 column
    const long row  = (long)blockIdx.x * ROWS_PB + r;
    const bool live = (row < Nl);
    const long rowc = live ? row : 0;            // clamp: keep EXEC full

    // 1) kick off async NT stage of the streaming local row into LDS
    stage16_nt(lf + rowc * C + c * 4, &tile[tid * 4]);

    // 2) meanwhile: voxel lookup + indirect gather from L2-resident gfeat
    const int bx = lb[0], by = lb[1], bz = lb[2];
    const int ox = gb[0], oy = gb[1], oz = gb[2];
    int gx = ((lc[rowc * 3 + 0] + bx) >> SCALE_SHIFT) + ox;
    int gy = ((lc[rowc * 3 + 1] + by) >> SCALE_SHIFT) + oy;
    int gz = ((lc[rowc * 3 + 2] + bz) >> SCALE_SHIFT) + oz;
    bool inb = (gx >= 0) & (gx < GRID) & (gy >= 0) & (gy < GRID) &
               (gz >= 0) & (gz < GRID);
    int cx = min(max(gx, 0), GRID - 1);
    int cy = min(max(gy, 0), GRID - 1);
    int cz = min(max(gz, 0), GRID - 1);
    int idx = map[(cx * GRID + cy) * GRID + cz];
    bool valid = inb & (idx >= 0);
    long grow = (long)max(idx, 0);

    // RT-policy gather: rows of gfeat stay hot in L2
    v4f gv = *(const v4f*)(gfeat + grow * C + c * 4);        // global_load_b128
    if (!valid) gv = (v4f)(0.f);

    // 3) consume the staged local row (own slot -> own-wave ASYNCcnt wait only)
    wait_async();
    v4f lv = *(const v4f*)&tile[tid * 4];                    // ds_load_b128

    v4f o = lv + gv;

    if (live)                                                // TH=NT b128 store
        __builtin_nontemporal_store(o, (v4f*)(out + row * C + c * 4));
}

// ---------------------------------------------------------------------------
extern "C" void kernel_launch(void* const* d_in, const int* in_sizes, int n_in,
                              void* d_out, int out_size, void* d_ws, size_t ws_size,
                              hipStream_t stream) {
    const float* lf    = (const float*)d_in[0];
    const int*   lc    = (const int*)  d_in[1];
    const int*   lb    = (const int*)  d_in[2];
    const float* gfeat = (const float*)d_in[3];
    const int*   gc    = (const int*)  d_in[4];
    const int*   gb    = (const int*)  d_in[5];
    float*       out   = (float*)d_out;
    int*         map   = (int*)d_ws;             // 32^3 * 4 B = 128 KB

    const int Nl = in_sizes[1] / 3;              // 200000
    const int Ng = in_sizes[4] / 3;              // 20000

    gf_init_map<<<(GRID3 + 255) / 256, 256, 0, stream>>>(map);
    gf_scatter_map<<<(Ng + 255) / 256, 256, 0, stream>>>(gc, map, Ng);
    gf_fuse<<<(Nl + ROWS_PB - 1) / ROWS_PB, THREADS, 0, stream>>>(
        lf, lc, lb, gfeat, gb, map, out, Nl);
}